// CapacityAwareRouter_41936060678275
// MI455X (gfx1250) — compile-verified
//
#include <hip/hip_runtime.h>
#include <hip/hip_bf16.h>

typedef __attribute__((ext_vector_type(2))) float v2f;
typedef __attribute__((ext_vector_type(8))) float v8f;

#define B_T    8192
#define DIM    2048
#define NEXP   64
#define TOPK   4

// ---------------------------------------------------------------------------
// Kernel 1: logits = x @ W^T + bias   via V_WMMA_F32_16X16X4_F32
// One wave computes a 16(tokens) x 64(experts) tile: 4 accumulators (v8f).
// A layout (16x4 f32, ISA 7.12.2): lane l<16 holds A[l][0..1], lane l+16 holds A[l][2..3].
// B layout (4x16 f32, mirrored):   lane l<16 holds B[0..1][l], lane l+16 holds B[2..3][l].
// x is read exactly once (64 MB) -> HBM-roofline optimal.
// ---------------------------------------------------------------------------
__global__ void router_gemm_wmma(const float* __restrict__ x,
                                 const float* __restrict__ W,
                                 const float* __restrict__ bias,
                                 float* __restrict__ logits) {
    const int wave = (int)((blockIdx.x * blockDim.x + threadIdx.x) >> 5); // 0..511
    const int lane = (int)(threadIdx.x & 31);
    const int m0   = wave * 16;
    const int mrow = m0 + (lane & 15);
    const int koff = (lane >> 4) * 2;          // 0 or 2

    v8f acc0 = (v8f)0.0f, acc1 = (v8f)0.0f, acc2 = (v8f)0.0f, acc3 = (v8f)0.0f;

    const float* __restrict__ xrow = x + (size_t)mrow * DIM + koff;
    const int nl = lane & 15;
    const float* __restrict__ w0 = W + (size_t)(nl)      * DIM + koff;
    const float* __restrict__ w1 = W + (size_t)(nl + 16) * DIM + koff;
    const float* __restrict__ w2 = W + (size_t)(nl + 32) * DIM + koff;
    const float* __restrict__ w3 = W + (size_t)(nl + 48) * DIM + koff;

    for (int k = 0; k < DIM; k += 4) {
        v2f a; a.x = xrow[k]; a.y = xrow[k + 1];
        v2f b0; b0.x = w0[k]; b0.y = w0[k + 1];
        v2f b1; b1.x = w1[k]; b1.y = w1[k + 1];
        v2f b2; b2.x = w2[k]; b2.y = w2[k + 1];
        v2f b3; b3.x = w3[k]; b3.y = w3[k + 1];
        acc0 = __builtin_amdgcn_wmma_f32_16x16x4_f32(false, a, false, b0, (short)0, acc0, false, false);
        acc1 = __builtin_amdgcn_wmma_f32_16x16x4_f32(false, a, false, b1, (short)0, acc1, false, false);
        acc2 = __builtin_amdgcn_wmma_f32_16x16x4_f32(false, a, false, b2, (short)0, acc2, false, false);
        acc3 = __builtin_amdgcn_wmma_f32_16x16x4_f32(false, a, false, b3, (short)0, acc3, false, false);
    }

    // C/D layout: VGPR r holds M = r + 8*(lane>=16), N = lane&15 (within tile)
    const int mbase = m0 + ((lane >> 4) * 8);
    {
        const int n = nl;
        const float bs = bias[n];
        #pragma unroll
        for (int r = 0; r < 8; ++r)
            logits[(size_t)(mbase + r) * NEXP + n] = acc0[r] + bs;
    }
    {
        const int n = nl + 16;
        const float bs = bias[n];
        #pragma unroll
        for (int r = 0; r < 8; ++r)
            logits[(size_t)(mbase + r) * NEXP + n] = acc1[r] + bs;
    }
    {
        const int n = nl + 32;
        const float bs = bias[n];
        #pragma unroll
        for (int r = 0; r < 8; ++r)
            logits[(size_t)(mbase + r) * NEXP + n] = acc2[r] + bs;
    }
    {
        const int n = nl + 48;
        const float bs = bias[n];
        #pragma unroll
        for (int r = 0; r < 8; ++r)
            logits[(size_t)(mbase + r) * NEXP + n] = acc3[r] + bs;
    }
}

// ---------------------------------------------------------------------------
// Kernel 2: scores = softmax(logits) per token (64 experts). One wave32/token,
// 2 experts per lane, shuffle-butterfly max + sum.
// ---------------------------------------------------------------------------
__global__ void softmax64(const float* __restrict__ logits,
                          float* __restrict__ scores) {
    const int wave = (int)((blockIdx.x * blockDim.x + threadIdx.x) >> 5);
    const int lane = (int)(threadIdx.x & 31);
    if (wave >= B_T) return;
    const float* rowp = logits + (size_t)wave * NEXP;
    float v0 = rowp[lane];
    float v1 = rowp[lane + 32];
    float m = fmaxf(v0, v1);
    #pragma unroll
    for (int s = 16; s >= 1; s >>= 1)
        m = fmaxf(m, __shfl_xor(m, s, 32));
    float e0 = __expf(v0 - m);
    float e1 = __expf(v1 - m);
    float sum = e0 + e1;
    #pragma unroll
    for (int s = 16; s >= 1; s >>= 1)
        sum += __shfl_xor(sum, s, 32);
    float inv = 1.0f / sum;
    scores[(size_t)wave * NEXP + lane]      = e0 * inv;
    scores[(size_t)wave * NEXP + lane + 32] = e1 * inv;
}

// ---------------------------------------------------------------------------
// Kernel 3: serial greedy capacity-aware routing. Single wave32, lane owns
// experts {lane, lane+32}. 4*8192 sequential steps (k-major, token-minor),
// mutating the logits rows in scratch exactly like the reference.
// Software-pipelined: the next token's row is prefetched into registers while
// the current row's argmax reduce runs, so the L2 load latency is off the
// serial critical path. (Row b+1 is never the row being stored, including at
// pass boundaries, so the prefetch is always safe.)
// ---------------------------------------------------------------------------
__global__ void route_serial(float* __restrict__ logits,
                             const float* __restrict__ scores,
                             const int* __restrict__ cap_ptr,
                             float* __restrict__ out_sel,
                             float* __restrict__ wraw) {
    const int lane = (int)threadIdx.x;          // 0..31, single wave
    const float NEG = -__builtin_inff();
    const float cap = (float)(cap_ptr[0] / TOPK);
    float rem0 = cap, rem1 = cap;               // remaining capacity, experts lane / lane+32

    // prologue: prefetch row 0
    float p0 = logits[lane];
    float p1 = logits[lane + 32];

    for (int i = 0; i < TOPK * B_T; ++i) {
        const int b     = i & (B_T - 1);
        const int kpass = i >> 13;              // B_T == 2^13

        float row0 = p0;
        float row1 = p1;

        // prefetch next step's row (always a different row than b)
        const int nb = (i + 1) & (B_T - 1);
        const float* nrowp = logits + (size_t)nb * NEXP;
        p0 = nrowp[lane];
        p1 = nrowp[lane + 32];

        float m0 = (rem0 > 0.0f) ? row0 : NEG;
        float m1 = (rem1 > 0.0f) ? row1 : NEG;

        // local pair, lowest-index tie-break (lane < lane+32)
        float bv; int bi;
        if (m1 > m0) { bv = m1; bi = lane + 32; } else { bv = m0; bi = lane; }
        // butterfly argmax, ties -> lowest expert index (matches jnp.argmax)
        #pragma unroll
        for (int s = 16; s >= 1; s >>= 1) {
            float ov = __shfl_xor(bv, s, 32);
            int   oi = __shfl_xor(bi, s, 32);
            if (ov > bv || (ov == bv && oi < bi)) { bv = ov; bi = oi; }
        }
        const int chosen = bi;
        const int src = chosen & 31;

        float rsel = (chosen < 32) ? rem0 : rem1;
        float rC   = __shfl(rsel, src, 32);     // remaining[chosen]
        float vsel = (chosen < 32) ? row0 : row1;
        float cval = __shfl(vsel, src, 32);     // row[chosen]
        const bool ok = (rC > 0.0f);

        if (ok) {
            // full experts strictly above the chosen logit get zapped
            if (rem0 <= 0.0f && row0 > cval) row0 = NEG;
            if (rem1 <= 0.0f && row1 > cval) row1 = NEG;
            if (chosen == lane)            rem0 -= 1.0f;
            else if (chosen == lane + 32)  rem1 -= 1.0f;
        } else {
            row0 = NEG;
            row1 = NEG;
        }
        float* rowp = logits + (size_t)b * NEXP;
        rowp[lane]      = row0;
        rowp[lane + 32] = row1;

        if (lane == 0) {
            out_sel[(size_t)b * TOPK + kpass] = ok ? (float)chosen : -1.0f;
            wraw[(size_t)b * TOPK + kpass]    = ok ? scores[(size_t)b * NEXP + chosen] : 0.0f;
        }
    }
}

// ---------------------------------------------------------------------------
// Kernel 4: normalize weights: w /= (sum + 1e-8), write into d_out (2nd half)
// ---------------------------------------------------------------------------
__global__ void normalize_weights(const float* __restrict__ wraw,
                                  float* __restrict__ out_w) {
    const int t = (int)(blockIdx.x * blockDim.x + threadIdx.x);
    if (t >= B_T) return;
    float a = wraw[(size_t)t * TOPK + 0];
    float b = wraw[(size_t)t * TOPK + 1];
    float c = wraw[(size_t)t * TOPK + 2];
    float d = wraw[(size_t)t * TOPK + 3];
    float inv = 1.0f / (a + b + c + d + 1e-8f);
    out_w[(size_t)t * TOPK + 0] = a * inv;
    out_w[(size_t)t * TOPK + 1] = b * inv;
    out_w[(size_t)t * TOPK + 2] = c * inv;
    out_w[(size_t)t * TOPK + 3] = d * inv;
}

extern "C" void kernel_launch(void* const* d_in, const int* in_sizes, int n_in,
                              void* d_out, int out_size, void* d_ws, size_t ws_size,
                              hipStream_t stream) {
    (void)in_sizes; (void)n_in; (void)out_size; (void)ws_size;

    const float* x    = (const float*)d_in[0];   // (8192, 2048)
    const float* W    = (const float*)d_in[1];   // (64, 2048)
    const float* bias = (const float*)d_in[2];   // (64,)
    const int*   cap  = (const int*)d_in[3];     // scalar token_capacity

    float* ws      = (float*)d_ws;
    float* logits  = ws;                          // 8192*64 floats (mutated by routing)
    float* scores  = ws + (size_t)B_T * NEXP;     // 8192*64 floats
    float* wraw    = ws + (size_t)2 * B_T * NEXP; // 8192*4 floats

    float* out_sel = (float*)d_out;               // selected as floats, (8192,4)
    float* out_w   = (float*)d_out + (size_t)B_T * TOPK; // normalized weights

    // 1) GEMM: 512 m-tiles, 8 waves/block -> 64 blocks of 256 threads
    router_gemm_wmma<<<64, 256, 0, stream>>>(x, W, bias, logits);

    // 2) softmax: 8192 waves -> 1024 blocks of 256 threads
    softmax64<<<1024, 256, 0, stream>>>(logits, scores);

    // 3) serial greedy routing: single wave32, software-pipelined row loads
    route_serial<<<1, 32, 0, stream>>>(logits, scores, cap, out_sel, wraw);

    // 4) normalize weights
    normalize_weights<<<32, 256, 0, stream>>>(wraw, out_w);
}